// GCNLayer_65403761983574
// MI455X (gfx1250) — compile-verified
//
#include <hip/hip_runtime.h>

typedef __attribute__((ext_vector_type(2))) float v2f;
typedef __attribute__((ext_vector_type(8))) float v8f;

// ---------------------------------------------------------------------------
// Phase 1: T[N, F_OUT] = X[N, F_IN] @ W[F_OUT, F_IN]^T + b
// One wave -> one 16x16 output tile, K-loop of V_WMMA_F32_16X16X4_F32.
// Block = 256 threads = 8 waves = 8 N-tiles (F_OUT = 128) of one 16-row stripe.
// ---------------------------------------------------------------------------
__global__ __launch_bounds__(256) void gcn_gemm_bias_wmma(
    const float* __restrict__ X,     // [N, F_IN]
    const float* __restrict__ W,     // [F_OUT, F_IN]
    const float* __restrict__ bias,  // [F_OUT]
    float* __restrict__ T,           // [N, F_OUT]
    int N, int F_IN, int F_OUT)
{
    const int lane  = threadIdx.x & 31;
    const int wave  = threadIdx.x >> 5;
    const int mBase = blockIdx.x * 16;
    const int nBase = wave * 16;
    if (nBase >= F_OUT) return;           // uniform per wave

    const int half = lane >> 4;           // 0: lanes 0-15, 1: lanes 16-31
    const int l15  = lane & 15;

    // A: 16x4 f32 tile of X.  Lane l: M = l&15, holds K = k + 2*half .. +1
    const int mRow = mBase + l15;
    const int mClamped = (mRow < N) ? mRow : (N - 1);
    const float* __restrict__ aRow = X + (size_t)mClamped * F_IN + half * 2;
    // B: 4x16 tile of W^T, i.e. B[k][n] = W[n][k].  Lane l: N = l&15.
    const int nRow = nBase + l15;
    const float* __restrict__ bRow = W + (size_t)nRow * F_IN + half * 2;

    v8f c = {0.f, 0.f, 0.f, 0.f, 0.f, 0.f, 0.f, 0.f};
    for (int k = 0; k < F_IN; k += 4) {
        v2f a = *(const v2f*)(aRow + k);
        v2f b = *(const v2f*)(bRow + k);
        // (neg_a, A, neg_b, B, c_mod, C, reuse_a, reuse_b)
        c = __builtin_amdgcn_wmma_f32_16x16x4_f32(
                false, a, false, b, (short)0, c, false, false);
    }

    const float bn = bias[nRow];
    // C/D layout: VGPR i -> M = mBase + 8*half + i, N = nBase + (lane&15)
    const int mOut = mBase + half * 8;
#pragma unroll
    for (int i = 0; i < 8; ++i) {
        const int m = mOut + i;
        if (m < N) T[(size_t)m * F_OUT + nRow] = c[i] + bn;
    }
}

// ---------------------------------------------------------------------------
// Zero-fill output (harness poisons d_out with 0xAA before timing).
// ---------------------------------------------------------------------------
__global__ __launch_bounds__(256) void gcn_zero(float* __restrict__ p, size_t n)
{
    size_t i = (size_t)blockIdx.x * blockDim.x + threadIdx.x;
    if (i < n) p[i] = 0.0f;
}

// ---------------------------------------------------------------------------
// Phase 2: SpMM scatter.  One wave per edge: coalesced 512B gather of
// T[col[e]], scale by val[e], 4 f32 atomic-adds per lane into out[row[e]].
// T (51.2 MB) and out (51.2 MB) are both L2-resident (192 MB L2).
// ---------------------------------------------------------------------------
__global__ __launch_bounds__(256) void gcn_spmm_scatter(
    const float* __restrict__ T,     // [N, F_OUT]
    const int*   __restrict__ rows,  // [E]
    const int*   __restrict__ cols,  // [E]
    const float* __restrict__ vals,  // [E]
    float* __restrict__ out,         // [N, F_OUT]
    int E, int F_OUT)
{
    const int lane = threadIdx.x & 31;
    const int wave = threadIdx.x >> 5;
    const int e = blockIdx.x * 8 + wave;
    if (e >= E) return;

    const int   r = rows[e];
    const int   cidx = cols[e];
    const float v = vals[e];

    // F_OUT = 128 -> 32 lanes x 4 floats (global_load_b128 per lane)
    const float4* src = (const float4*)(T + (size_t)cidx * F_OUT) + lane;
    float*        dst = out + (size_t)r * F_OUT + lane * 4;

    float4 t = *src;
    atomicAdd(dst + 0, t.x * v);
    atomicAdd(dst + 1, t.y * v);
    atomicAdd(dst + 2, t.z * v);
    atomicAdd(dst + 3, t.w * v);
}

// ---------------------------------------------------------------------------
extern "C" void kernel_launch(void* const* d_in, const int* in_sizes, int n_in,
                              void* d_out, int out_size, void* d_ws, size_t ws_size,
                              hipStream_t stream)
{
    const float* X    = (const float*)d_in[0];  // [N, F_IN]
    const int*   rows = (const int*)  d_in[1];  // [E]
    const int*   cols = (const int*)  d_in[2];  // [E]
    const float* vals = (const float*)d_in[3];  // [E]
    const float* W    = (const float*)d_in[4];  // [F_OUT, F_IN]
    const float* b    = (const float*)d_in[5];  // [F_OUT]

    const int F_OUT = in_sizes[5];              // 128
    const int F_IN  = in_sizes[4] / F_OUT;      // 256
    const int N     = in_sizes[0] / F_IN;       // 100000
    const int E     = in_sizes[1];              // 1600000

    float* T   = (float*)d_ws;                  // [N, F_OUT] scratch (51.2 MB)
    float* out = (float*)d_out;                 // [N, F_OUT]

    // 1) Dense GEMM + bias via WMMA f32
    const int mTiles = (N + 15) / 16;           // 6250
    gcn_gemm_bias_wmma<<<mTiles, 256, 0, stream>>>(X, W, b, T, N, F_IN, F_OUT);

    // 2) Zero the output accumulator
    const size_t outN = (size_t)N * F_OUT;
    gcn_zero<<<(int)((outN + 255) / 256), 256, 0, stream>>>(out, outN);

    // 3) Edge scatter with f32 atomics
    gcn_spmm_scatter<<<(E + 7) / 8, 256, 0, stream>>>(T, rows, cols, vals, out, E, F_OUT);
}